// Sampler_14688788152879
// MI455X (gfx1250) — compile-verified
//
#include <hip/hip_runtime.h>
#include <stdint.h>

#define NT 1024
#define NB1 2048          // level-1 bins: key bits 30:20
#define NB2 2048          // level-2 bins: key bits 19:9
#define NB3 512           // level-3 bins: key bits 8:0 (exact value)
#define NOCAP 0x7FFFFFFFu
#define TILE (NT * 2)     // async staging tile: 2 floats per lane (b64)

// Block-wide exclusive scan of (uint count, float mass) pairs across NT threads.
// Also returns grand totals. Wave32-aware (shfl_up width 32, 32 wave partials).
__device__ __forceinline__ void scan2(uint32_t c, float s, uint32_t tid,
                                      uint32_t* sc, float* ss,
                                      uint32_t& Cexcl, float& Sexcl,
                                      uint32_t& Ctot, float& Stot) {
  __syncthreads();
  uint32_t lane = tid & 31u, w = tid >> 5;
  uint32_t ci = c; float si = s;
#pragma unroll
  for (int o = 1; o < 32; o <<= 1) {
    uint32_t c2 = __shfl_up(ci, o, 32);
    float    s2 = __shfl_up(si, o, 32);
    if (lane >= (uint32_t)o) { ci += c2; si += s2; }
  }
  if (lane == 31u) { sc[w] = ci; ss[w] = si; }
  __syncthreads();
  if (w == 0u) {
    uint32_t cw = sc[lane]; float sw = ss[lane];
#pragma unroll
    for (int o = 1; o < 32; o <<= 1) {
      uint32_t c2 = __shfl_up(cw, o, 32);
      float    s2 = __shfl_up(sw, o, 32);
      if (lane >= (uint32_t)o) { cw += c2; sw += s2; }
    }
    sc[lane] = cw; ss[lane] = sw;
  }
  __syncthreads();
  uint32_t cb = (w > 0) ? sc[w - 1] : 0u;
  float    sb = (w > 0) ? ss[w - 1] : 0.f;
  Cexcl = cb + (ci - c);
  Sexcl = sb + (si - s);
  Ctot = sc[31]; Stot = ss[31];
  __syncthreads();
}

extern "C" __global__ void __launch_bounds__(NT)
sampler_full(const float* __restrict__ logits,
             const float* __restrict__ temps,
             const int*   __restrict__ topks,
             const float* __restrict__ topps,
             const float* __restrict__ minps,
             const float* __restrict__ uvec,
             int*   __restrict__ tok_out,
             float* __restrict__ prob_out,
             int V) {
  __shared__ uint32_t h1c[NB1]; __shared__ float h1s[NB1];
  __shared__ uint32_t h2c[NB2]; __shared__ float h2s[NB2];
  __shared__ float stage[2][TILE];                // ping-pong async staging (16KB)
  __shared__ float redm[32]; __shared__ float reds[32];
  __shared__ uint32_t sc[32]; __shared__ float ss[32];
  __shared__ uint32_t s_pos; __shared__ uint32_t s_C; __shared__ float s_S;
  __shared__ uint32_t s_Ct; __shared__ float s_St;
  __shared__ float sMs, sInvT, sInvDen, sTopP, sThr, sTotal, sTarget, sInvTot;
  __shared__ uint32_t sK, sThrKey;
  __shared__ uint32_t s_bin1, s_pre2, s_cutKey, s_nv, s_sampKey;
  __shared__ uint32_t s_posCut1, s_posCut2; __shared__ int s_exh2;
  __shared__ uint32_t s_C0; __shared__ float s_S0;
  __shared__ uint32_t s_binS1, s_preS2; __shared__ float s_SeS;
  __shared__ int s_jsel; __shared__ int s_tok;

  const uint32_t tid = threadIdx.x;
  const int b = blockIdx.x;
  const float* __restrict__ row = logits + (size_t)b * (size_t)V;

  if (tid == 0) {
    sInvT = 1.0f / temps[b];
    sTopP = topps[b];
    sK = (uint32_t)topks[b];
  }
  __syncthreads();
  const float invT = sInvT;

  // ---- P1: fused online softmax (max + sum-exp), vectorized b128 loads ----
  {
    float m = -3.4e38f, s = 0.f;
    for (int i = (int)tid * 4; i < V; i += NT * 4) {
      if (i + NT * 16 < V) __builtin_prefetch(row + i + NT * 16, 0, 0);
      if (i + 3 < V) {
        float4 l4 = *(const float4*)(row + i);
        float pl[4] = { l4.x, l4.y, l4.z, l4.w };
#pragma unroll
        for (int j = 0; j < 4; j++) {
          float x = pl[j] * invT;
          if (x > m) { s = s * expf(m - x) + 1.f; m = x; }
          else       { s += expf(x - m); }
        }
      } else {
        for (int t = i; t < V; t++) {
          float x = row[t] * invT;
          if (x > m) { s = s * expf(m - x) + 1.f; m = x; }
          else       { s += expf(x - m); }
        }
      }
    }
#pragma unroll
    for (int o = 16; o > 0; o >>= 1) {
      float mo = __shfl_xor(m, o, 32);
      float so = __shfl_xor(s, o, 32);
      float nm = fmaxf(m, mo);
      s = s * expf(m - nm) + so * expf(mo - nm);
      m = nm;
    }
    if ((tid & 31u) == 0u) { redm[tid >> 5] = m; reds[tid >> 5] = s; }
    __syncthreads();
    if (tid < 32) {
      float mm = redm[tid], ssum = reds[tid];
#pragma unroll
      for (int o = 16; o > 0; o >>= 1) {
        float mo = __shfl_xor(mm, o, 32);
        float so = __shfl_xor(ssum, o, 32);
        float nm = fmaxf(mm, mo);
        ssum = ssum * expf(mm - nm) + so * expf(mo - nm);
        mm = nm;
      }
      if (tid == 0) {
        sMs = mm;
        float inv = 1.f / ssum;
        sInvDen = inv;
        float thr = minps[b] * inv;     // p_max * min_p (p_max = 1/denom)
        sThr = thr; sThrKey = __float_as_uint(thr);
      }
    }
    __syncthreads();
  }
  const float Ms = sMs;
  const float invDen = sInvDen;

  auto probAt = [&](int i) -> float {
    return expf(row[i] * invT - Ms) * invDen;
  };

  // ---- P3: level-1 histogram; double-buffered async global->LDS staging ----
  for (int i = tid; i < NB1; i += NT) { h1c[i] = 0u; h1s[i] = 0.f; }
  __syncthreads();
  {
    const int nTiles = (V + TILE - 1) / TILE;
    auto issueTile = [&](int t) {
      int i2 = t * TILE + (int)tid * 2;
      if (i2 + 1 < V) {
        uint32_t loff = (uint32_t)(uintptr_t)(&stage[t & 1][tid * 2]);
        uint64_t ga = (uint64_t)(uintptr_t)(row + i2);
        asm volatile("global_load_async_to_lds_b64 %0, %1, off"
                     :: "v"(loff), "v"(ga) : "memory");
      }
    };
    issueTile(0);
    for (int t = 0; t < nTiles; t++) {
      if (t + 1 < nTiles) issueTile(t + 1);             // overlap DMA with compute
      asm volatile("s_wait_asynccnt 0x1" ::: "memory"); // oldest tile retired
      __syncthreads();
      int i2 = t * TILE + (int)tid * 2;
      bool full = (i2 + 1 < V);
#pragma unroll
      for (int j = 0; j < 2; j++) {
        int i = i2 + j;
        if (i < V) {
          float l = full ? stage[t & 1][tid * 2 + j] : row[i];
          float p = expf(l * invT - Ms) * invDen;
          uint32_t key = __float_as_uint(p);
          uint32_t bin = key >> 20;
          atomicAdd(&h1c[bin], 1u);
          atomicAdd(&h1s[bin], p);
        }
      }
      __syncthreads();
    }
    asm volatile("s_wait_asynccnt 0x0" ::: "memory");
  }

  // Vectorized filtered-histogram builder: b128 loads, L2-resident passes.
  auto histPass = [&](uint32_t shift, uint32_t prefix, uint32_t mask,
                      uint32_t* hc, float* hs) {
    for (int i = (int)tid * 4; i < V; i += NT * 4) {
      if (i + 3 < V) {
        float4 l4 = *(const float4*)(row + i);
        float pl[4] = { l4.x, l4.y, l4.z, l4.w };
#pragma unroll
        for (int j = 0; j < 4; j++) {
          float p = expf(pl[j] * invT - Ms) * invDen;
          uint32_t key = __float_as_uint(p);
          if ((key >> shift) == prefix) {
            uint32_t bn = (shift == 20u) ? ((key >> 9) & 0x7FFu) : (key & mask);
            atomicAdd(&hc[bn], 1u); atomicAdd(&hs[bn], p);
          }
        }
      } else {
        for (int t = i; t < V; t++) {
          float p = probAt(t);
          uint32_t key = __float_as_uint(p);
          if ((key >> shift) == prefix) {
            uint32_t bn = (shift == 20u) ? ((key >> 9) & 0x7FFu) : (key & mask);
            atomicAdd(&hc[bn], 1u); atomicAdd(&hs[bn], p);
          }
        }
      }
    }
  };

  // Generic descending-bin walk over 2048 bins (2 positions/thread).
  auto walk = [&](const uint32_t* hc, const float* hs,
                  uint32_t kk, float pp, uint32_t cap,
                  uint32_t Cb, float Sb) {
    if (tid == 0) s_pos = NOCAP;
    uint32_t p0 = tid * 2u;
    uint32_t c0 = hc[2047u - p0], c1 = hc[2046u - p0];
    float    m0 = hs[2047u - p0], m1 = hs[2046u - p0];
    uint32_t Ce; float Se; uint32_t Ct; float St;
    scan2(c0 + c1, m0 + m1, tid, sc, ss, Ce, Se, Ct, St);
    Ce += Cb; Se += Sb;
    uint32_t hit = NOCAP, hC = 0; float hS = 0.f;
    if ((Ce + c0 >= kk) || (Se + m0 > pp) || (p0 >= cap)) { hit = p0; hC = Ce; hS = Se; }
    else {
      uint32_t Ce1 = Ce + c0; float Se1 = Se + m0;
      if ((Ce1 + c1 >= kk) || (Se1 + m1 > pp) || (p0 + 1u >= cap)) { hit = p0 + 1u; hC = Ce1; hS = Se1; }
    }
    if (tid == NT - 1) { s_Ct = Cb + Ct; s_St = Sb + St; }
    if (hit != NOCAP) atomicMin(&s_pos, hit);
    __syncthreads();
    if (hit != NOCAP && hit == s_pos) { s_C = hC; s_S = hS; }
    __syncthreads();
  };

  // ---- W1: level-1 cutoff walk (min over top-k / top-p / min-p crossings) ----
  const uint32_t thrBin1 = sThrKey >> 20;
  walk(h1c, h1s, sK, sTopP, (NB1 - 1u) - thrBin1, 0u, 0.f);
  if (tid == 0) { s_posCut1 = s_pos; s_bin1 = 2047u - s_pos; s_C0 = s_C; s_S0 = s_S; }
  __syncthreads();
  const uint32_t bin1 = s_bin1;

  // ---- P4/W2: level-2 cutoff refinement ----
  for (int i = tid; i < NB2; i += NT) { h2c[i] = 0u; h2s[i] = 0.f; }
  __syncthreads();
  histPass(20u, bin1, 0x7FFu, h2c, h2s);
  __syncthreads();
  uint32_t cap2 = (thrBin1 == bin1) ? ((NB2 - 1u) - ((sThrKey >> 9) & 0x7FFu)) : NOCAP;
  walk(h2c, h2s, sK, sTopP, cap2, s_C0, s_S0);
  if (tid == 0) {
    if (s_pos == NOCAP) { // whole bin kept; cutoff just below it
      s_exh2 = 1;
      if (bin1 == 0u) { s_cutKey = 0u; s_nv = 0x7FFFFFFFu; }
      else { s_cutKey = (bin1 << 20) - 1u; s_nv = 0u; }
      sTotal = s_St;
    } else {
      s_exh2 = 0; s_posCut2 = s_pos;
      uint32_t b2 = 2047u - s_pos;
      s_pre2 = (bin1 << 11) | b2;
      s_C0 = s_C; s_S0 = s_S;
    }
  }
  __syncthreads();

  // ---- P5/W3: level-3 exact-value resolution of cutoff ----
  if (!s_exh2) {
    for (int i = tid; i < NB3; i += NT) { h2c[i] = 0u; h2s[i] = 0.f; }
    __syncthreads();
    histPass(9u, s_pre2, 0x1FFu, h2c, h2s);
    __syncthreads();
    if (tid == 0) {
      uint32_t Ce = s_C0; float Se = s_S0;
      uint32_t kk = sK; float pp = sTopP, thr = sThr;
      int found = 0;
      for (int p = 0; p < NB3; p++) {
        uint32_t bin = (uint32_t)(NB3 - 1 - p);
        uint32_t c = h2c[bin]; float s = h2s[bin];
        uint32_t key = (s_pre2 << 9) | bin; float v = __uint_as_float(key);
        long long nk = (long long)kk - (long long)Ce;
        long long np;
        if (Se > pp) np = 0;
        else if (v > 0.f) { float q = (pp - Se) / v; np = (q > 1.0e9f) ? (1LL << 40) : ((long long)q + 1); }
        else np = (1LL << 40);
        long long nm = (v >= thr) ? (1LL << 40) : 0;
        long long n = nk; if (np < n) n = np; if (nm < n) n = nm;
        if (n < (long long)c || n <= 0) {
          long long nv = (n < 0) ? 0 : n;
          s_cutKey = key; s_nv = (uint32_t)nv;
          sTotal = Se + (float)nv * v;
          found = 1; break;
        } else { Ce += c; Se += s; }
      }
      if (!found) {
        uint32_t basek = (s_pre2 << 9);
        if (basek == 0u) { s_cutKey = 0u; s_nv = 0x7FFFFFFFu; }
        else { s_cutKey = basek - 1u; s_nv = 0u; }
        sTotal = Se;
      }
    }
  }
  __syncthreads();
  if (tid == 0) { sInvTot = 1.f / sTotal; sTarget = uvec[b] * sTotal; }
  __syncthreads();

  // ---- W1b: level-1 sampling walk (mass crossing at target, capped at cutoff) ----
  walk(h1c, h1s, 0xFFFFFFFFu, sTarget, s_posCut1, 0u, 0.f);
  if (tid == 0) { s_binS1 = 2047u - s_pos; s_SeS = s_S; }
  __syncthreads();
  const uint32_t binS1 = s_binS1;

  // ---- P6/W4: level-2 sampling refinement ----
  for (int i = tid; i < NB2; i += NT) { h2c[i] = 0u; h2s[i] = 0.f; }
  __syncthreads();
  histPass(20u, binS1, 0x7FFu, h2c, h2s);
  __syncthreads();
  uint32_t capS2 = (binS1 == bin1 && !s_exh2) ? s_posCut2 : NOCAP;
  walk(h2c, h2s, 0xFFFFFFFFu, sTarget, capS2, 0u, s_SeS);
  if (tid == 0) {
    uint32_t ps = (s_pos == NOCAP) ? 2047u : s_pos;
    s_preS2 = (binS1 << 11) | (2047u - ps);
    s_SeS = (s_pos == NOCAP) ? s_St : s_S;
  }
  __syncthreads();

  // ---- P7/W5: level-3 sampling: exact value + index-rank within tie group ----
  for (int i = tid; i < NB3; i += NT) { h2c[i] = 0u; h2s[i] = 0.f; }
  __syncthreads();
  histPass(9u, s_preS2, 0x1FFu, h2c, h2s);
  __syncthreads();
  uint32_t preS2 = s_preS2;
  if (tid == 0) {
    float Se = s_SeS; float tgt = sTarget;
    uint32_t lastKey = s_cutKey; uint32_t lastC = 1; int got = 0;
    for (int p = 0; p < NB3 && !got; p++) {
      uint32_t bin = (uint32_t)(NB3 - 1 - p);
      uint32_t c = h2c[bin]; if (!c) continue;
      float s = h2s[bin];
      uint32_t key = (preS2 << 9) | bin; float v = __uint_as_float(key);
      lastKey = key; lastC = c;
      if ((Se + s > tgt) || (key <= s_cutKey)) {
        float q = tgt - Se;
        long long jj = (v > 0.f) ? (long long)(q / v) : 0;
        if (jj < 0) jj = 0;
        if (jj > (long long)c - 1) jj = (long long)c - 1;
        if (key == s_cutKey && jj > (long long)s_nv - 1) jj = (long long)s_nv - 1;
        if (jj < 0) jj = 0;
        s_sampKey = key; s_jsel = (int)jj; got = 1;
      } else Se += s;
    }
    if (!got) { s_sampKey = lastKey; int j = (int)lastC - 1; s_jsel = (j < 0) ? 0 : j; }
    s_tok = 0;
  }
  __syncthreads();

  // ---- P8/P9: tie index-ranks (stable order) + write outputs + pick token ----
  const uint32_t cutKey = s_cutKey, sampKey = s_sampKey, nv = s_nv;
  const int jsel = s_jsel;
  const float invTot = sInvTot;
  const int chunk = (((V + NT - 1) / NT) + 3) & ~3;   // multiple of 4 -> aligned b128
  const int i0 = (int)tid * chunk;
  const int i1 = min(i0 + chunk, V);
  uint32_t c1 = 0, c2 = 0;
  for (int i = i0; i < i1; i += 4) {
    if (i + 3 < i1) {
      float4 l4 = *(const float4*)(row + i);
      float pl[4] = { l4.x, l4.y, l4.z, l4.w };
#pragma unroll
      for (int j = 0; j < 4; j++) {
        uint32_t key = __float_as_uint(expf(pl[j] * invT - Ms) * invDen);
        c1 += (key == cutKey); c2 += (key == sampKey);
      }
    } else {
      for (int t = i; t < i1; t++) {
        uint32_t key = __float_as_uint(probAt(t));
        c1 += (key == cutKey); c2 += (key == sampKey);
      }
    }
  }
  uint32_t Ce; float Se; uint32_t Ct; float St;
  scan2(c1, (float)c2, tid, sc, ss, Ce, Se, Ct, St);
  uint32_t r1 = Ce;
  uint32_t r2 = (uint32_t)(Se + 0.5f);   // counts <= 2^24: exact in float
  float* __restrict__ orow = prob_out + (size_t)b * (size_t)V;
  for (int i = i0; i < i1; i += 4) {
    if (i + 3 < i1) {
      float4 l4 = *(const float4*)(row + i);
      float pl[4] = { l4.x, l4.y, l4.z, l4.w };
      float o4[4];
#pragma unroll
      for (int j = 0; j < 4; j++) {
        float p = expf(pl[j] * invT - Ms) * invDen;
        uint32_t key = __float_as_uint(p);
        float o = 0.f;
        if (key > cutKey) o = p * invTot;
        else if (key == cutKey) { if (r1 < nv) o = p * invTot; r1++; }
        o4[j] = o;
        if (key == sampKey) { if ((int)r2 == jsel) s_tok = i + j; r2++; }
      }
      *(float4*)(orow + i) = make_float4(o4[0], o4[1], o4[2], o4[3]);
    } else {
      for (int t = i; t < i1; t++) {
        float p = probAt(t);
        uint32_t key = __float_as_uint(p);
        float o = 0.f;
        if (key > cutKey) o = p * invTot;
        else if (key == cutKey) { if (r1 < nv) o = p * invTot; r1++; }
        orow[t] = o;
        if (key == sampKey) { if ((int)r2 == jsel) s_tok = t; r2++; }
      }
    }
  }
  __syncthreads();
  if (tid == 0) tok_out[b] = s_tok;
}

extern "C" void kernel_launch(void* const* d_in, const int* in_sizes, int n_in,
                              void* d_out, int out_size, void* d_ws, size_t ws_size,
                              hipStream_t stream) {
  const float* logits = (const float*)d_in[0];
  const float* temps  = (const float*)d_in[1];
  const int*   topks  = (const int*)d_in[2];
  const float* topps  = (const float*)d_in[3];
  const float* minps  = (const float*)d_in[4];
  const float* u      = (const float*)d_in[5];
  const int B = in_sizes[1];
  const int V = in_sizes[0] / B;
  int*   tok   = (int*)d_out;
  float* probs = (float*)d_out + B;
  hipLaunchKernelGGL(sampler_full, dim3(B), dim3(NT), 0, stream,
                     logits, temps, topks, topps, minps, u, tok, probs, V);
}